// HashEncoder_2D_11192684774176
// MI455X (gfx1250) — compile-verified
//
#include <hip/hip_runtime.h>

// ---- hash-grid meta (mirrors reference _level_meta) ----
// res:    16, 32, 64, 128, 256, 512, 1024, 2048
// size:   4096, 32768, 262144, then 524288 for levels 3..7 (all powers of two)
// dense (under_hash) for levels 0..2, instant-ngp fast_hash for levels 3..7
#define NLEVELS 8

typedef float v4f __attribute__((ext_vector_type(4)));

__global__ __launch_bounds__(256) void hashgrid_encode_kernel(
    const float* __restrict__ positions,   // [B,3]
    const float* __restrict__ table,       // [TOTAL_ENTRIES,2]
    float* __restrict__ out,               // [B,16]
    int B)
{
    const int b = blockIdx.x * blockDim.x + threadIdx.x;
    if (b >= B) return;

    // Streaming read of this point's position: non-temporal (don't pollute L2,
    // keep the 23MB hash table resident there). Compiler merges to b96 NT.
    const float px = __builtin_nontemporal_load(positions + 3 * b + 0);
    const float py = __builtin_nontemporal_load(positions + 3 * b + 1);
    const float pz = __builtin_nontemporal_load(positions + 3 * b + 2);

    // per-level table sizes (powers of two) and cumulative offsets
    constexpr unsigned kSize[NLEVELS] = {
        4096u, 32768u, 262144u, 524288u, 524288u, 524288u, 524288u, 524288u};
    constexpr unsigned kOff[NLEVELS] = {
        0u, 4096u, 36864u, 299008u, 823296u, 1347584u, 1871872u, 2396160u};

    const float2* __restrict__ tab2 = (const float2*)table;

    float acc[2 * NLEVELS];

#pragma unroll
    for (int lvl = 0; lvl < NLEVELS; ++lvl) {
        const unsigned res   = 16u << lvl;         // 16*2^lvl
        const float    scale = (float)(res - 1u);  // 16*2^lvl - 1 (exact in f32)

        const float fx = px * scale + 0.5f;
        const float fy = py * scale + 0.5f;
        const float fz = pz * scale + 0.5f;
        const float gx = floorf(fx);
        const float gy = floorf(fy);
        const float gz = floorf(fz);
        const float rx = fx - gx;   // frac
        const float ry = fy - gy;
        const float rz = fz - gz;
        const unsigned ux = (unsigned)gx;
        const unsigned uy = (unsigned)gy;
        const unsigned uz = (unsigned)gz;

        const float wx[2] = {1.0f - rx, rx};
        const float wy[2] = {1.0f - ry, ry};
        const float wz[2] = {1.0f - rz, rz};

        const unsigned mask = kSize[lvl] - 1u;
        const unsigned off  = kOff[lvl];

        float s0 = 0.0f, s1 = 0.0f;

#pragma unroll
        for (int c = 0; c < 8; ++c) {
            const unsigned bx = (unsigned)(c) & 1u;
            const unsigned by = (unsigned)(c >> 1) & 1u;
            const unsigned bz = (unsigned)(c >> 2) & 1u;
            const unsigned cx = ux + bx;
            const unsigned cy = uy + by;
            const unsigned cz = uz + bz;

            unsigned index;
            if (lvl < 3) {
                // dense row-major (uint32 wrap), then power-of-two modulo
                index = cx + cy * res + cz * (res * res);
            } else {
                // instant-ngp spatial hash (uint32 wrap)
                index = cx ^ (cy * 2654435761u) ^ (cz * 805459861u);
            }
            index &= mask;

            // temporal load: table stays hot in the 192MB L2
            const float2 e = tab2[off + index];

            const float w = wx[bx] * wy[by] * wz[bz];
            s0 = fmaf(w, e.x, s0);
            s1 = fmaf(w, e.y, s1);
        }

        acc[2 * lvl + 0] = s0;
        acc[2 * lvl + 1] = s1;
    }

    // Streaming output: 64B per point, 64B-aligned. Non-temporal b128 stores
    // so the 128MB of results do not evict the hash table from L2.
    float* orow = out + 16 * (size_t)b;
#pragma unroll
    for (int q = 0; q < 4; ++q) {
        v4f v;
        v.x = acc[4 * q + 0];
        v.y = acc[4 * q + 1];
        v.z = acc[4 * q + 2];
        v.w = acc[4 * q + 3];
        __builtin_nontemporal_store(v, (v4f*)(orow + 4 * q));
    }
}

extern "C" void kernel_launch(void* const* d_in, const int* in_sizes, int n_in,
                              void* d_out, int out_size, void* d_ws, size_t ws_size,
                              hipStream_t stream) {
    const float* positions = (const float*)d_in[0];   // [B,3] f32
    const float* table     = (const float*)d_in[1];   // [TOTAL_ENTRIES,2] f32
    float* out             = (float*)d_out;           // [B,16] f32

    const int B = in_sizes[0] / 3;
    const int threads = 256;                          // 8 wave32 per block
    const int blocks  = (B + threads - 1) / threads;

    hashgrid_encode_kernel<<<blocks, threads, 0, stream>>>(positions, table, out, B);
}